// NLCCSCriterion_69492570849635
// MI455X (gfx1250) — compile-verified
//
#include <hip/hip_runtime.h>
#include <hip/hip_bf16.h>
#include <stdint.h>

// NLCCS criterion for MI455X (gfx1250).
// One workgroup (1024 threads = 32 waves) per 256x256 mask.
// Entire label image (256 KB) lives in LDS (CDNA5: 320 KB/WGP).
// Tile is DMA'd into LDS with the Tensor Data Mover (TENSORcnt),
// then converted in place to labels. CCL = 8-neighbor min propagation
// + 1-hop pointer jumping until fixpoint (same fixpoint as reference).
// All sweeps are vectorized 4 pixels/thread with ds_load_b128/ds_store_b128.

#define HH 256
#define WW 256
#define HWPIX 65536
#define SENTL 0x7FFFFFFFu
#define NT 1024

typedef unsigned int v4u __attribute__((ext_vector_type(4)));
typedef int          v8i __attribute__((ext_vector_type(8)));
typedef int          v4i __attribute__((ext_vector_type(4)));
typedef float        v4f __attribute__((ext_vector_type(4)));

// Issue a 2D TENSOR_LOAD_TO_LDS of a 256x256 f32 tile at gbase -> LDS offset.
// D# per CDNA5 ISA ch.8: group0 = {count/lds_addr/global_addr/type},
// group1 = {data_size=4B, tensor_dim0/1=256, tile_dim0/1=256, stride0=256}.
__device__ __forceinline__ void tdm_load_tile_f32(const float* gbase,
                                                  unsigned lds_byte_off) {
  unsigned long long ga = (unsigned long long)(uintptr_t)gbase;
  v4u g0;
  g0[0] = 1u;                                            // count=1 (valid), user bits 0
  g0[1] = lds_byte_off;                                  // lds_addr
  g0[2] = (unsigned)(ga & 0xFFFFFFFFull);                // global_addr[31:0]
  g0[3] = (unsigned)((ga >> 32) & 0x01FFFFFFull)         // global_addr[56:32]
        | (2u << 30);                                    // type=2 ("image")
  v8i g1;
  g1[0] = (int)(2u << 16);      // workgroup_mask=0 (not in cluster), data_size=2 (4B)
  g1[1] = (int)(256u << 16);    // tensor_dim0[15:0]=256 in bits[63:48]
  g1[2] = (int)(256u << 16);    // tensor_dim0 hi=0; tensor_dim1[15:0]=256
  g1[3] = (int)(256u << 16);    // tensor_dim1 hi=0; tile_dim0=256
  g1[4] = 256;                  // tile_dim1=256; tile_dim2=0
  g1[5] = 256;                  // tensor_dim0_stride[31:0]=256
  g1[6] = 0;                    // stride0 hi=0; tensor_dim1_stride lo=0
  g1[7] = 0;
  v4i gz = (v4i){0, 0, 0, 0};
#if __clang_major__ >= 23
  v8i gz8 = (v8i){0, 0, 0, 0, 0, 0, 0, 0};
  __builtin_amdgcn_tensor_load_to_lds(g0, g1, gz, gz, gz8, 0);
#else
  __builtin_amdgcn_tensor_load_to_lds(g0, g1, gz, gz, 0);
#endif
  __builtin_amdgcn_s_wait_tensorcnt(0);
}

__global__ __launch_bounds__(NT) void nlccs_mask_kernel(
    const float* __restrict__ masks, float* __restrict__ part) {
  extern __shared__ __align__(16) unsigned int lab[];   // 65536 u32 = 256 KB LDS
  __shared__ unsigned s_flag, s_bg, s_h0, s_w0, s_h1p, s_w1p;
  __shared__ unsigned long long s_k1, s_k2;
  __shared__ float s_part[NT / 32];

  const int tid  = (int)threadIdx.x;
  const int wid  = tid >> 5;
  const int lane = tid & 31;
  const float* base = masks + (size_t)blockIdx.x * HWPIX;

  if (tid == 0) {
    s_bg = 0u; s_k1 = 0ull; s_k2 = 0ull;
    s_h0 = HH; s_w0 = WW; s_h1p = 0u; s_w1p = 0u;
  }
  if (wid == 0) {
    // wave-level DMA: low 32 bits of a flat LDS pointer == LDS byte offset
    tdm_load_tile_f32(base, (unsigned)(uintptr_t)(void*)lab);
  }
  __syncthreads();

  // --- init labels in place: fg <=> raw mask value > 0 (b128 in/out) ---
  for (int i = tid * 4; i < HWPIX; i += NT * 4) {
    v4u x = *(const v4u*)(lab + i);
    v4u o;
#pragma unroll
    for (int k = 0; k < 4; ++k)
      o[k] = (__uint_as_float(x[k]) > 0.0f) ? (unsigned)(i + k) : SENTL;
    *(v4u*)(lab + i) = o;
  }
  __syncthreads();

  // --- CCL: 8-neighbor min + pointer jump, 4 pixels/thread per sweep ---
  for (int it = 0; it < 1024; ++it) {
    if (tid == 0) s_flag = 0u;
    __syncthreads();
    unsigned changed = 0u;
    for (int i = tid * 4; i < HWPIX; i += NT * 4) {
      v4u v = *(const v4u*)(lab + i);
      if (v[0] == SENTL && v[1] == SENTL && v[2] == SENTL && v[3] == SENTL)
        continue;
      const int r  = i >> 8;
      const int c0 = i & 255;
      unsigned t[6], b[6];
#pragma unroll
      for (int k = 0; k < 6; ++k) { t[k] = SENTL; b[k] = SENTL; }
      if (r > 0) {
        const int j = i - WW;
        v4u tv = *(const v4u*)(lab + j);
        t[1] = tv[0]; t[2] = tv[1]; t[3] = tv[2]; t[4] = tv[3];
        if (c0 > 0)      t[0] = lab[j - 1];
        if (c0 + 4 < WW) t[5] = lab[j + 4];
      }
      if (r < HH - 1) {
        const int j = i + WW;
        v4u bv = *(const v4u*)(lab + j);
        b[1] = bv[0]; b[2] = bv[1]; b[3] = bv[2]; b[4] = bv[3];
        if (c0 > 0)      b[0] = lab[j - 1];
        if (c0 + 4 < WW) b[5] = lab[j + 4];
      }
      unsigned hv[6];
      hv[0] = (c0 > 0) ? lab[i - 1] : SENTL;
      hv[1] = v[0]; hv[2] = v[1]; hv[3] = v[2]; hv[4] = v[3];
      hv[5] = (c0 + 4 < WW) ? lab[i + 4] : SENTL;
      v4u nv = v;
      unsigned any = 0u;
#pragma unroll
      for (int k = 0; k < 4; ++k) {
        const unsigned vk = v[k];
        if (vk == SENTL) continue;
        unsigned m = vk;
        m = min(m, t[k]);     m = min(m, t[k + 1]); m = min(m, t[k + 2]);
        m = min(m, b[k]);     m = min(m, b[k + 1]); m = min(m, b[k + 2]);
        m = min(m, hv[k]);    m = min(m, hv[k + 2]);
        const unsigned pj = lab[m];   // labels only hold fg indices -> valid
        if (pj < m) m = pj;           // one pointer-jump hop
        if (m < vk) { nv[k] = m; any = 1u; }
      }
      if (any) { *(v4u*)(lab + i) = nv; changed = 1u; }
    }
    if (changed) s_flag = 1u;
    __syncthreads();
    const unsigned f = s_flag;
    __syncthreads();                   // protect read vs next-iter reset
    if (!f) break;
  }

  // --- areas: pack count into upper 16 bits of each root's slot.
  //     Merge equal-root runs inside each 4-vector before the ds_add. ---
  {
    unsigned bgl = 0u;
    for (int i = tid * 4; i < HWPIX; i += NT * 4) {
      v4u v = *(const v4u*)(lab + i);
      unsigned roots[4]; unsigned cnt[4]; int nseg = 0;
#pragma unroll
      for (int k = 0; k < 4; ++k) {
        const unsigned vk = v[k];
        if (vk == SENTL) { bgl++; continue; }
        const unsigned rt = vk & 0xFFFFu;
        if (nseg > 0 && roots[nseg - 1] == rt) cnt[nseg - 1]++;
        else { roots[nseg] = rt; cnt[nseg] = 1u; nseg++; }
      }
      for (int s = 0; s < nseg; ++s)
        atomicAdd(&lab[roots[s]], cnt[s] << 16);  // low 16 bits stay = root
    }
    if (bgl) atomicAdd(&s_bg, bgl);
    __syncthreads();
  }

  // --- top-2 by (area, lower-label tie-break); include background bin ---
  for (int i = tid * 4; i < HWPIX; i += NT * 4) {
    v4u v = *(const v4u*)(lab + i);
#pragma unroll
    for (int k = 0; k < 4; ++k) {
      const unsigned vk = v[k];
      if (vk != SENTL && (vk & 0xFFFFu) == (unsigned)(i + k)) {  // root slot
        unsigned long long key = ((unsigned long long)(vk >> 16) << 32) |
                                 (unsigned long long)(0xFFFFFFFFu - (unsigned)(i + k));
        atomicMax(&s_k1, key);
      }
    }
  }
  if (tid == 0) {
    unsigned long long bgkey = ((unsigned long long)s_bg << 32) |
                               (unsigned long long)(0xFFFFFFFFu - 65536u);
    atomicMax(&s_k1, bgkey);
  }
  __syncthreads();
  const unsigned long long k1 = s_k1;
  for (int i = tid * 4; i < HWPIX; i += NT * 4) {
    v4u v = *(const v4u*)(lab + i);
#pragma unroll
    for (int k = 0; k < 4; ++k) {
      const unsigned vk = v[k];
      if (vk != SENTL && (vk & 0xFFFFu) == (unsigned)(i + k)) {
        unsigned long long key = ((unsigned long long)(vk >> 16) << 32) |
                                 (unsigned long long)(0xFFFFFFFFu - (unsigned)(i + k));
        if (key != k1) atomicMax(&s_k2, key);
      }
    }
  }
  if (tid == 0) {
    unsigned long long bgkey = ((unsigned long long)s_bg << 32) |
                               (unsigned long long)(0xFFFFFFFFu - 65536u);
    if (bgkey != k1) atomicMax(&s_k2, bgkey);
  }
  __syncthreads();

  // --- bbox of second-largest component (thread-local, then 4 atomics) ---
  const unsigned k2lab = 0xFFFFFFFFu - (unsigned)(s_k2 & 0xFFFFFFFFull);
  {
    unsigned lh0 = HH, lw0 = WW, lh1 = 0u, lw1 = 0u;
    for (int i = tid * 4; i < HWPIX; i += NT * 4) {
      v4u v = *(const v4u*)(lab + i);
      const unsigned r  = (unsigned)i >> 8;
      const unsigned c0 = (unsigned)i & 255u;
#pragma unroll
      for (int k = 0; k < 4; ++k) {
        const unsigned vk = v[k];
        const bool sel = (k2lab == 65536u)
                             ? (vk == SENTL)
                             : (vk != SENTL && (vk & 0xFFFFu) == k2lab);
        if (sel) {
          const unsigned c = c0 + (unsigned)k;
          lh0 = min(lh0, r); lh1 = max(lh1, r + 1u);
          lw0 = min(lw0, c); lw1 = max(lw1, c + 1u);
        }
      }
    }
    if (lh1 != 0u) {
      atomicMin(&s_h0, lh0);  atomicMax(&s_h1p, lh1);
      atomicMin(&s_w0, lw0);  atomicMax(&s_w1p, lw1);
    }
    __syncthreads();
  }

  // --- masked mean of m = max((x+1)/2, 0), zeroed inside bbox ---
  const unsigned h0 = s_h0, h1 = s_h1p, w0 = s_w0, w1 = s_w1p;
  float acc = 0.0f;
  for (int i = tid * 4; i < HWPIX; i += NT * 4) {
    v4f x = *(const v4f*)(base + i);
    const unsigned r  = (unsigned)i >> 8;
    const unsigned c0 = (unsigned)i & 255u;
    const bool rowIn = (r >= h0) && (r < h1);
#pragma unroll
    for (int k = 0; k < 4; ++k) {
      const float mv = fmaxf((x[k] + 1.0f) * 0.5f, 0.0f);
      const unsigned c = c0 + (unsigned)k;
      const bool inside = rowIn && (c >= w0) && (c < w1);
      acc += inside ? 0.0f : mv;
    }
  }
  for (int off = 16; off > 0; off >>= 1) acc += __shfl_down(acc, off, 32);
  if (lane == 0) s_part[wid] = acc;
  __syncthreads();
  if (wid == 0) {
    float t = s_part[lane];
    for (int off = 16; off > 0; off >>= 1) t += __shfl_down(t, off, 32);
    if (lane == 0) part[blockIdx.x] = t * (1.0f / (float)HWPIX);
  }
}

__global__ void nlccs_finalize(const float* __restrict__ part,
                               float* __restrict__ out, int n) {
  __shared__ float sp[4];
  const int tid = (int)threadIdx.x;   // 128 threads, n == 128
  float v = (tid < n) ? part[tid] : 0.0f;
  for (int off = 16; off > 0; off >>= 1) v += __shfl_down(v, off, 32);
  if ((tid & 31) == 0) sp[tid >> 5] = v;
  __syncthreads();
  if (tid == 0) out[0] = (sp[0] + sp[1] + sp[2] + sp[3]) / (float)n;
}

extern "C" void kernel_launch(void* const* d_in, const int* in_sizes, int n_in,
                              void* d_out, int out_size, void* d_ws, size_t ws_size,
                              hipStream_t stream) {
  (void)n_in; (void)out_size; (void)ws_size;
  const float* masks = (const float*)d_in[0];
  float* part = (float*)d_ws;                 // 128 floats of scratch
  const int nmask = in_sizes[0] / HWPIX;      // B*K = 128
  nlccs_mask_kernel<<<dim3(nmask), dim3(NT), HWPIX * sizeof(unsigned), stream>>>(
      masks, part);
  nlccs_finalize<<<dim3(1), dim3(128), 0, stream>>>(part, (float*)d_out, nmask);
}